// MultiScaledConvHead_81028853006751
// MI455X (gfx1250) — compile-verified
//
#include <hip/hip_runtime.h>
#include <hip/hip_bf16.h>
#include <math.h>
#include <stdint.h>

// Problem constants (from the reference)
constexpr int kB  = 4;
constexpr int kS  = 2048;
constexpr int kD  = 512;
constexpr int kH  = 8;
constexpr int kM  = 4;
constexpr int kR  = 32;
constexpr int kHD = 64;   // D / H
constexpr float kEPS = 1e-8f;

typedef __attribute__((ext_vector_type(2))) float v2f;
typedef __attribute__((ext_vector_type(8))) float v8f;

// CDNA5 fp32 WMMA: D(16x16,f32) = A(16x4,f32) x B(4x16,f32) + C
// A frag (2 VGPR): lane L holds A[L%16][2*(L/16) + i]   (i = vgpr index)
// B frag (2 VGPR): lane L holds B[2*(L/16) + i][L%16]
// C/D (8 VGPR):    vgpr v, lane L holds C[v + 8*(L/16)][L%16]
__device__ __forceinline__ v8f wmma4(v2f a, v2f b, v8f c) {
    return __builtin_amdgcn_wmma_f32_16x16x4_f32(
        /*neg_a=*/false, a, /*neg_b=*/false, b,
        /*c_mod=*/(short)0, c, /*reuse_a=*/false, /*reuse_b=*/false);
}

// Async global->LDS copy, 16B per lane (GLOBAL_LOAD_ASYNC_TO_LDS_B128, GVS
// addressing: mem = SGPR base + VGPR 32-bit offset; LDS addr from VGPR).
// Tracked by ASYNCcnt; data bypasses VGPRs entirely.
__device__ __forceinline__ void async_b128_to_lds(uint32_t lds_addr, uint32_t goff,
                                                  uint64_t sbase) {
    asm volatile("global_load_async_to_lds_b128 %0, %1, %2"
                 :: "v"(lds_addr), "v"(goff), "s"(sbase)
                 : "memory");
}
__device__ __forceinline__ void wait_asynccnt0() {
    asm volatile("s_wait_asynccnt 0" ::: "memory");
}

// ---------------------------------------------------------------------------
// Kernel 1: Q/K/V projections.  out = x @ W.T + b, scattered to [B,H,S,HD].
// One wave -> 16x64 output tile. 3*512*8 = 12288 waves.
// ---------------------------------------------------------------------------
__global__ __launch_bounds__(256) void k_qkv(
    const float* __restrict__ x,
    const float* __restrict__ Wq, const float* __restrict__ bq,
    const float* __restrict__ Wk, const float* __restrict__ bk,
    const float* __restrict__ Wv, const float* __restrict__ bv,
    float* __restrict__ q, float* __restrict__ k, float* __restrict__ v)
{
    const int wave  = (blockIdx.x * blockDim.x + threadIdx.x) >> 5;
    const int lane  = threadIdx.x & 31;
    const int which = wave / 4096;          // 0=q, 1=k, 2=v
    const int t     = wave - which * 4096;
    const int row0  = (t >> 3) << 4;        // 512 row-tiles of 16 (over B*S)
    const int col0  = (t & 7) << 6;         // 8 col-tiles of 64  (over D)

    const float* W    = (which == 0) ? Wq : (which == 1) ? Wk : Wv;
    const float* bias = (which == 0) ? bq : (which == 1) ? bk : bv;
    float*       out  = (which == 0) ? q  : (which == 1) ? k  : v;

    const int l16   = lane & 15;
    const int khalf = (lane >> 4) << 1;     // 0 or 2

    v8f acc[4] = {};
    const float* xrow = x + (size_t)(row0 + l16) * kD + khalf;
    const float* wrow = W + (size_t)(col0 + l16) * kD + khalf;

    for (int kk = 0; kk < kD; kk += 4) {
        v2f a = *(const v2f*)(xrow + kk);
#pragma unroll
        for (int nt = 0; nt < 4; ++nt) {
            v2f b = *(const v2f*)(wrow + (size_t)nt * 16 * kD + kk);
            acc[nt] = wmma4(a, b, acc[nt]);
        }
    }

    const int crow = (lane >> 4) << 3;      // 0 or 8
#pragma unroll
    for (int nt = 0; nt < 4; ++nt) {
#pragma unroll
        for (int vv = 0; vv < 8; ++vv) {
            const int i  = row0 + crow + vv;      // global row over B*S
            const int j  = col0 + nt * 16 + l16;  // global col over D
            const int b_ = i >> 11;               // / S
            const int s_ = i & (kS - 1);
            const int h_ = j >> 6;                // / HD
            const int hd = j & (kHD - 1);
            out[(((size_t)(b_ * kH + h_)) * kS + s_) * kHD + hd] = acc[nt][vv] + bias[j];
        }
    }
}

// ---------------------------------------------------------------------------
// Kernel 2: lm[m,s,t] = sum_r mask1[m,r,s] * mask2[m,r,t].
// GEMM per m: A[s][r] = mask1[m][r][s] (transposed read), B = mask2[m].
// One wave -> 16x64 tile. 4 * 128 * 32 = 16384 waves.
// ---------------------------------------------------------------------------
__global__ __launch_bounds__(256) void k_lm(
    const float* __restrict__ mask1, const float* __restrict__ mask2,
    float* __restrict__ lm)
{
    const int wave = (blockIdx.x * blockDim.x + threadIdx.x) >> 5;
    const int lane = threadIdx.x & 31;
    const int m    = wave / 4096;
    const int t    = wave - m * 4096;
    const int s0   = (t >> 5) << 4;         // 128 s-tiles of 16
    const int t0   = (t & 31) << 6;         // 32 t-tiles of 64

    const int l16   = lane & 15;
    const int khalf = (lane >> 4) << 1;

    const float* m1 = mask1 + (size_t)m * kR * kS;
    const float* m2 = mask2 + (size_t)m * kR * kS;

    v8f acc[4] = {};
#pragma unroll
    for (int kk = 0; kk < kR; kk += 4) {
        const int kb = kk + khalf;
        v2f a;
        a[0] = m1[(size_t)kb * kS + s0 + l16];
        a[1] = m1[(size_t)(kb + 1) * kS + s0 + l16];
#pragma unroll
        for (int nt = 0; nt < 4; ++nt) {
            v2f b;
            b[0] = m2[(size_t)kb * kS + t0 + nt * 16 + l16];
            b[1] = m2[(size_t)(kb + 1) * kS + t0 + nt * 16 + l16];
            acc[nt] = wmma4(a, b, acc[nt]);
        }
    }

    const int crow = (lane >> 4) << 3;
#pragma unroll
    for (int nt = 0; nt < 4; ++nt)
#pragma unroll
        for (int vv = 0; vv < 8; ++vv)
            lm[((size_t)m * kS + s0 + crow + vv) * kS + t0 + nt * 16 + l16] = acc[nt][vv];
}

// ---------------------------------------------------------------------------
// Kernel 3: flash-style pass. One BLOCK = one (b,h) and 128 s-rows; each of
// its 8 waves owns a 16-row slice. K/V 16x64 tiles are double-buffered in LDS
// via GLOBAL_LOAD_ASYNC_TO_LDS_B128 (ASYNCcnt), shared by all 8 waves:
//   att16x16 = (q kT)/8 * lm[m, s, roll[s,t]]   (gather applied in registers)
//   ssq    += sum(att^2)          (Frobenius accumulator, scalar per (b,h))
//   O16x64 += att16x16 @ V16x64   (att bounced through LDS for A-layout)
// Frobenius scale is applied later (scalar per (b,h), commutes with att@v).
// ---------------------------------------------------------------------------
__global__ __launch_bounds__(256) void k_attn(
    const float* __restrict__ q, const float* __restrict__ k,
    const float* __restrict__ v, const float* __restrict__ lm,
    const int* __restrict__ roll,
    float* __restrict__ ouns, float* __restrict__ partial)
{
    __shared__ __align__(16) float kvbuf[2][2][16 * 64]; // [buf][K/V][row*64+hd] 16KB
    __shared__ __align__(16) float att_lds[8][256];      // 16x16 bounce tile/wave 8KB

    const int tid  = threadIdx.x;
    const int widb = tid >> 5;
    const int lane = tid & 31;

    const int bh   = blockIdx.x >> 4;        // b*H + h   (32 total)
    const int sblk = blockIdx.x & 15;        // 16 s-superblocks of 128 rows
    const int s0   = sblk * 128 + widb * 16; // this wave's 16 rows
    const int mm   = bh & (kM - 1);          // h % M  (H multiple of M)

    const float* qb  = q  + (size_t)bh * kS * kHD;
    const float* lmm = lm + (size_t)mm * kS * kS;
    const uint64_t kbase = (uint64_t)(uintptr_t)(k + (size_t)bh * kS * kHD);
    const uint64_t vbase = (uint64_t)(uintptr_t)(v + (size_t)bh * kS * kHD);

    const int l16   = lane & 15;
    const int khalf = (lane >> 4) << 1;
    const int crow  = (lane >> 4) << 3;

    // Cooperative staging: 256 threads x 16B cover one 16x64 f32 tile.
    const uint32_t strow = (uint32_t)(tid >> 4);       // 0..15
    const uint32_t stcol = (uint32_t)((tid & 15) * 4); // 0..60, 16B aligned
    const uint32_t ldsK  = (uint32_t)(uintptr_t)&kvbuf[0][0][strow * 64 + stcol];
    const uint32_t ldsV  = (uint32_t)(uintptr_t)&kvbuf[0][1][strow * 64 + stcol];
    const uint32_t bufStride = (uint32_t)(sizeof(float) * 2 * 16 * 64); // 8KB
#define STAGE(BUF, T0)                                                        \
    do {                                                                      \
        const uint32_t goff = (((uint32_t)(T0) + strow) * kHD + stcol) * 4u;  \
        async_b128_to_lds(ldsK + (BUF) * bufStride, goff, kbase);             \
        async_b128_to_lds(ldsV + (BUF) * bufStride, goff, vbase);             \
    } while (0)

    // Preload the 16x64 Q block as 16 A-fragments (kept in VGPRs).
    v2f qa[16];
#pragma unroll
    for (int kk = 0; kk < 16; ++kk)
        qa[kk] = *(const v2f*)(qb + (size_t)(s0 + l16) * kHD + 4 * kk + khalf);

    v8f oacc[4] = {};
    float ssql = 0.0f;
    float* tile = att_lds[widb];

    STAGE(0, 0);
    int cur = 0;
    for (int t0 = 0; t0 < kS; t0 += 16) {
        wait_asynccnt0();       // this wave's async stores to LDS landed
        __syncthreads();        // everyone's landed; prev buffer free to reuse
        if (t0 + 16 < kS) STAGE(cur ^ 1, t0 + 16);   // overlap next tile load

        const float* kt = kvbuf[cur][0];
        const float* vt = kvbuf[cur][1];

        // --- att tile = Q(16x64) @ K(16x64)^T  (K from LDS) ---
        v8f attc = {};
#pragma unroll
        for (int kk = 0; kk < 16; ++kk) {
            v2f bf = *(const v2f*)(kt + l16 * 64 + 4 * kk + khalf);
            attc = wmma4(qa[kk], bf, attc);
        }
        // --- elementwise: 1/sqrt(HD), gathered lm, ssq, stash to LDS ---
#pragma unroll
        for (int vv = 0; vv < 8; ++vv) {
            const int srow = s0 + crow + vv;
            const int tcol = t0 + l16;
            const int gidx = roll[(size_t)srow * kS + tcol];
            const float lmv = lmm[(size_t)srow * kS + gidx];
            const float a = attc[vv] * 0.125f * lmv;
            ssql += a * a;
            tile[(crow + vv) * 16 + l16] = a;
        }
        // same-wave LDS is in-order: store->load needs no barrier
        // --- O(16x64) += att(16x16) @ V(16x64)  (V from LDS) ---
#pragma unroll
        for (int kk = 0; kk < 4; ++kk) {
            v2f af = *(const v2f*)(&tile[l16 * 16 + 4 * kk + khalf]);
#pragma unroll
            for (int nt = 0; nt < 4; ++nt) {
                v2f bf;
                bf[0] = vt[(4 * kk + khalf) * 64 + nt * 16 + l16];
                bf[1] = vt[(4 * kk + khalf + 1) * 64 + nt * 16 + l16];
                oacc[nt] = wmma4(af, bf, oacc[nt]);
            }
        }
        cur ^= 1;
    }
#undef STAGE

    // wave-reduce ssq (wave32) and emit one deterministic partial per wave
#pragma unroll
    for (int off = 16; off > 0; off >>= 1)
        ssql += __shfl_xor(ssql, off, 32);
    if (lane == 0) partial[bh * 128 + sblk * 8 + widb] = ssql;

#pragma unroll
    for (int nt = 0; nt < 4; ++nt)
#pragma unroll
        for (int vv = 0; vv < 8; ++vv)
            ouns[(size_t)bh * kS * kHD + (size_t)(s0 + crow + vv) * kHD + nt * 16 + l16]
                = oacc[nt][vv];
}

// ---------------------------------------------------------------------------
// Kernel 4a: reduce per-wave ssq partials -> ssq[b*H+h]  (deterministic)
// ---------------------------------------------------------------------------
__global__ void k_ssqred(const float* __restrict__ partial, float* __restrict__ ssq)
{
    const int t = threadIdx.x;
    if (t < kB * kH) {
        float s = 0.0f;
        for (int i = 0; i < kS / 16; ++i) s += partial[t * (kS / 16) + i];
        ssq[t] = s;
    }
}

// ---------------------------------------------------------------------------
// Kernel 4b: out = (scale[b,h] * Ouns) @ Wo.T + bo, scale folded into A loads.
// One wave -> 16x64 tile. 512*8 = 4096 waves.
// ---------------------------------------------------------------------------
__global__ __launch_bounds__(256) void k_out(
    const float* __restrict__ ouns, const float* __restrict__ ssq,
    const float* __restrict__ Wo, const float* __restrict__ bo,
    float* __restrict__ out)
{
    const int wave = (blockIdx.x * blockDim.x + threadIdx.x) >> 5;
    const int lane = threadIdx.x & 31;
    const int row0 = (wave >> 3) << 4;      // over B*S
    const int col0 = (wave & 7) << 6;       // over D

    const int l16   = lane & 15;
    const int khalf = (lane >> 4) << 1;
    const int b_    = row0 >> 11;           // tile never straddles a batch
    const int s_    = (row0 + l16) & (kS - 1);

    v8f acc[4] = {};
    for (int d0 = 0; d0 < kD; d0 += 4) {
        const int d  = d0 + khalf;
        const int h_ = d >> 6;              // uniform across the k-step
        const float sc = 1.0f / (sqrtf(ssq[b_ * kH + h_]) + kEPS);
        const float* orow =
            ouns + (((size_t)(b_ * kH + h_)) * kS + s_) * kHD + (d & (kHD - 1));
        v2f a;
        a[0] = orow[0] * sc;
        a[1] = orow[1] * sc;
#pragma unroll
        for (int nt = 0; nt < 4; ++nt) {
            v2f b = *(const v2f*)(Wo + (size_t)(col0 + nt * 16 + l16) * kD + d);
            acc[nt] = wmma4(a, b, acc[nt]);
        }
    }

    const int crow = (lane >> 4) << 3;
#pragma unroll
    for (int nt = 0; nt < 4; ++nt)
#pragma unroll
        for (int vv = 0; vv < 8; ++vv) {
            const int i = row0 + crow + vv;
            const int j = col0 + nt * 16 + l16;
            out[(size_t)i * kD + j] = acc[nt][vv] + bo[j];
        }
}

// ---------------------------------------------------------------------------
extern "C" void kernel_launch(void* const* d_in, const int* in_sizes, int n_in,
                              void* d_out, int out_size, void* d_ws, size_t ws_size,
                              hipStream_t stream) {
    (void)in_sizes; (void)n_in; (void)out_size; (void)ws_size;

    const float* x     = (const float*)d_in[0];
    const float* Wq    = (const float*)d_in[1];
    const float* bq    = (const float*)d_in[2];
    const float* Wk    = (const float*)d_in[3];
    const float* bk    = (const float*)d_in[4];
    const float* Wv    = (const float*)d_in[5];
    const float* bv    = (const float*)d_in[6];
    const float* mask1 = (const float*)d_in[7];
    const float* mask2 = (const float*)d_in[8];
    const float* Wo    = (const float*)d_in[9];
    const float* bo    = (const float*)d_in[10];
    const int*   roll  = (const int*)d_in[11];

    const size_t nQKV = (size_t)kB * kH * kS * kHD;   // 4,194,304
    const size_t nLM  = (size_t)kM * kS * kS;         // 16,777,216
    float* ws      = (float*)d_ws;
    float* q       = ws;
    float* k       = q + nQKV;
    float* v       = k + nQKV;
    float* lm      = v + nQKV;
    float* ouns    = lm + nLM;
    float* partial = ouns + nQKV;                     // 4096 floats
    float* ssq     = partial + 4096;                  // 32 floats

    k_qkv  <<<1536, 256, 0, stream>>>(x, Wq, bq, Wk, bk, Wv, bv, q, k, v);
    k_lm   <<<2048, 256, 0, stream>>>(mask1, mask2, lm);
    k_attn <<<512,  256, 0, stream>>>(q, k, v, lm, roll, ouns, partial);
    k_ssqred<<<1,    32, 0, stream>>>(partial, ssq);
    k_out  <<<512,  256, 0, stream>>>(ouns, ssq, Wo, bo, (float*)d_out);
}